// memory_32512902431684
// MI455X (gfx1250) — compile-verified
//
#include <hip/hip_runtime.h>
#include <hip/hip_bf16.h>
#include <stdint.h>

// ---------------- problem constants (match reference) ----------------
#define BATCH 1024
#define MEM   65536
#define DIM   256
#define KTOP  128
#define BETA  1e-8f
#define ALPHA 0.5f

typedef float v2f __attribute__((ext_vector_type(2)));
typedef float v8f __attribute__((ext_vector_type(8)));

// ---------------- Kernel 1: fp32 WMMA GEMM -> scores -----------------
// scores[b][m] = exp(q[b].key[m] - 1) * (hist[m] + BETA)
// Block: 64 rows of q x 256 columns of memory. 256 threads = 8 waves.
// Each wave owns one 16-col N-tile at a time and computes 4 stacked
// 16x16 M-tiles against it -> 1 global B load feeds 4 wmma ops.

#define MT           4                    // M-tiles per wave
#define ROWS_PER_BLK (16 * MT)            // 64
#define COLS_PER_BLK 256
#define THREADS      256
#define QS_STRIDE    260                  // 260 % 64 == 4 -> conflict-free b64

__global__ __launch_bounds__(THREADS)
void scores_gemm_kernel(const float* __restrict__ q,
                        const float* __restrict__ mkey,
                        const float* __restrict__ mhist,
                        float* __restrict__ scores)
{
    __shared__ float qs[ROWS_PER_BLK][QS_STRIDE];   // ~66.5 KB of 320 KB/WGP

    const int tid     = threadIdx.x;
    const int rowBase = blockIdx.y * ROWS_PER_BLK;
    const int colBase = blockIdx.x * COLS_PER_BLK;

    // Stage the 64x256 q tile into LDS once per block.
    for (int i = tid; i < ROWS_PER_BLK * DIM; i += THREADS) {
        const int r = i >> 8;          // DIM == 256
        const int c = i & (DIM - 1);
        qs[r][c] = q[(size_t)(rowBase + r) * DIM + c];
    }
    __syncthreads();

    const int wave = tid >> 5;
    const int lane = tid & 31;
    const int half = lane >> 4;        // 0: K = d0,d0+1   1: K = d0+2,d0+3
    const int lr   = lane & 15;        // A: M row (mod 16)   B: N column

    for (int t = wave; t < COLS_PER_BLK / 16; t += THREADS / 32) {
        const int n0 = colBase + t * 16;
        const float* brow = mkey + (size_t)(n0 + lr) * DIM + 2 * half;
        const float* arow0 = &qs[ 0 + lr][2 * half];
        const float* arow1 = &qs[16 + lr][2 * half];
        const float* arow2 = &qs[32 + lr][2 * half];
        const float* arow3 = &qs[48 + lr][2 * half];

        v8f acc0 = {}, acc1 = {}, acc2 = {}, acc3 = {};
#pragma unroll 4
        for (int d0 = 0; d0 < DIM; d0 += 4) {
            // B fragment 4x16: lanes 0-15 hold {K=d0,d0+1}, lanes 16-31 {d0+2,d0+3}
            const v2f b = *(const v2f*)(brow + d0);
            // A fragments 16x4, one per M-tile, same K striping
            const v2f a0 = *(const v2f*)(arow0 + d0);
            const v2f a1 = *(const v2f*)(arow1 + d0);
            const v2f a2 = *(const v2f*)(arow2 + d0);
            const v2f a3 = *(const v2f*)(arow3 + d0);
            acc0 = __builtin_amdgcn_wmma_f32_16x16x4_f32(false, a0, false, b, (short)0, acc0, false, false);
            acc1 = __builtin_amdgcn_wmma_f32_16x16x4_f32(false, a1, false, b, (short)0, acc1, false, false);
            acc2 = __builtin_amdgcn_wmma_f32_16x16x4_f32(false, a2, false, b, (short)0, acc2, false, false);
            acc3 = __builtin_amdgcn_wmma_f32_16x16x4_f32(false, a3, false, b, (short)0, acc3, false, false);
        }

        // C layout: vgpr i, lanes 0-15 -> (M=i, N=lr); lanes 16-31 -> (M=8+i, N=lr)
        const int col = n0 + lr;
        const float hb = mhist[col] + BETA;
#pragma unroll
        for (int i = 0; i < 8; ++i) {
            const int row = rowBase + i + 8 * half;
            scores[(size_t)(row     ) * MEM + col] = expf(acc0[i] - 1.0f) * hb;
            scores[(size_t)(row + 16) * MEM + col] = expf(acc1[i] - 1.0f) * hb;
            scores[(size_t)(row + 32) * MEM + col] = expf(acc2[i] - 1.0f) * hb;
            scores[(size_t)(row + 48) * MEM + col] = expf(acc3[i] - 1.0f) * hb;
        }
    }
}

// ------------- Kernel 2: exact top-128 radix select + posterior -------------
// Scores are strictly positive floats -> bit pattern order == value order.
// 3-level radix histogram (12 + 12 + 8 bits) finds the exact 128th-largest
// value T and how many ties at T to keep (smallest indices first, matching
// jax.lax.top_k). Final pass accumulates v > T and collects ties in one scan.

#define NB0 4096

__device__ __forceinline__ void suffix_select(const unsigned* hist, int nbins,
                                              unsigned need,
                                              unsigned* sh_sel, unsigned* sh_need)
{
    unsigned acc = 0;
    for (int bin = nbins - 1; bin >= 0; --bin) {
        acc += hist[bin];
        if (acc >= need) {
            *sh_sel  = (unsigned)bin;
            *sh_need = need - (acc - hist[bin]);
            return;
        }
    }
    *sh_sel = 0; *sh_need = need;   // unreachable for valid input
}

__global__ __launch_bounds__(256)
void topk_posterior_kernel(const float* __restrict__ scores,
                           const float* __restrict__ mhist,
                           const float* __restrict__ mvals,
                           float* __restrict__ out)
{
    __shared__ unsigned hist[NB0];
    __shared__ unsigned sh_sel, sh_need;
    __shared__ int      tieIdx[256];
    __shared__ int      sh_tieCnt;
    __shared__ float    redA[256], redB[256];

    const int tid = threadIdx.x;
    const int b   = blockIdx.x;
    const float*    s  = scores + (size_t)b * MEM;
    const unsigned* su = (const unsigned*)s;

    unsigned need = KTOP;
    unsigned prefix = 0;

    // ---- level 0: bits [31:20] ----
    for (int i = tid; i < NB0; i += 256) hist[i] = 0;
    __syncthreads();
    for (int i = tid; i < MEM; i += 256) {
        atomicAdd(&hist[su[i] >> 20], 1u);
    }
    __syncthreads();
    if (tid == 0) suffix_select(hist, NB0, need, &sh_sel, &sh_need);
    __syncthreads();
    prefix = sh_sel; need = sh_need;
    __syncthreads();

    // ---- level 1: bits [19:8] among (v>>20)==prefix ----
    for (int i = tid; i < NB0; i += 256) hist[i] = 0;
    __syncthreads();
    for (int i = tid; i < MEM; i += 256) {
        const unsigned v = su[i];
        if ((v >> 20) == prefix) atomicAdd(&hist[(v >> 8) & 0xFFFu], 1u);
    }
    __syncthreads();
    if (tid == 0) suffix_select(hist, NB0, need, &sh_sel, &sh_need);
    __syncthreads();
    prefix = (prefix << 12) | sh_sel; need = sh_need;
    __syncthreads();

    // ---- level 2: bits [7:0] among (v>>8)==prefix ----
    if (tid < 256) hist[tid] = 0;
    __syncthreads();
    for (int i = tid; i < MEM; i += 256) {
        const unsigned v = su[i];
        if ((v >> 8) == prefix) atomicAdd(&hist[v & 0xFFu], 1u);
    }
    __syncthreads();
    if (tid == 0) { suffix_select(hist, 256, need, &sh_sel, &sh_need); sh_tieCnt = 0; }
    __syncthreads();
    const unsigned T = (prefix << 8) | sh_sel;
    const unsigned tieNeed = sh_need;   // how many elements == T to include
    __syncthreads();

    // ---- fused pass: accumulate v > T, collect tie indices (v == T) ----
    // joint = exp(sim-1)*(ALPHA*h+BETA) = score * (ALPHA*h+BETA)/(h+BETA)
    float sumJ = 0.0f, sumJV = 0.0f;
    for (int i = tid; i < MEM; i += 256) {
        const unsigned v = su[i];
        if (v > T) {
            const float h  = mhist[i];
            const float sc = __uint_as_float(v);
            const float joint = sc * (ALPHA * h + BETA) / (h + BETA);
            sumJ  += joint;
            sumJV += joint * mvals[i];
        } else if (v == T) {
            const int p = atomicAdd(&sh_tieCnt, 1);
            if (p < 256) tieIdx[p] = i;
        }
    }
    __syncthreads();

    // ---- include the tieNeed smallest-index ties (rare; list is tiny) ----
    const int tieCnt = sh_tieCnt < 256 ? sh_tieCnt : 256;
    for (int j = tid; j < tieCnt; j += 256) {
        const int idx = tieIdx[j];
        int rank = 0;
        for (int k2 = 0; k2 < tieCnt; ++k2) rank += (tieIdx[k2] < idx);
        if (rank < (int)tieNeed) {
            const float h  = mhist[idx];
            const float sc = s[idx];
            const float joint = sc * (ALPHA * h + BETA) / (h + BETA);
            sumJ  += joint;
            sumJV += joint * mvals[idx];
        }
    }

    redA[tid] = sumJ; redB[tid] = sumJV;
    __syncthreads();
    for (int off = 128; off > 0; off >>= 1) {
        if (tid < off) { redA[tid] += redA[tid + off]; redB[tid] += redB[tid + off]; }
        __syncthreads();
    }
    if (tid == 0) out[b] = redB[0] / redA[0];
}

// ------------------------------ launcher ------------------------------
extern "C" void kernel_launch(void* const* d_in, const int* in_sizes, int n_in,
                              void* d_out, int out_size, void* d_ws, size_t ws_size,
                              hipStream_t stream)
{
    const float* q     = (const float*)d_in[0];   // [1024, 256]
    const float* mkey  = (const float*)d_in[1];   // [65536, 256]
    const float* mhist = (const float*)d_in[2];   // [65536]
    const float* mvals = (const float*)d_in[3];   // [65536]
    float* out    = (float*)d_out;                // [1024]
    float* scores = (float*)d_ws;                 // needs BATCH*MEM*4 = 256 MB

    dim3 g1(MEM / COLS_PER_BLK, BATCH / ROWS_PER_BLK);   // (256, 16)
    scores_gemm_kernel<<<g1, THREADS, 0, stream>>>(q, mkey, mhist, scores);

    topk_posterior_kernel<<<BATCH, 256, 0, stream>>>(scores, mhist, mvals, out);
}